// AnchorTargetLayer_90486370992177
// MI455X (gfx1250) — compile-verified
//
#include <hip/hip_runtime.h>
#include <hip/hip_bf16.h>

// ---------------- static problem constants (from reference setup_inputs) ----
#define AB        9          // anchors per cell (3 ratios x 3 scales)
#define BATCH     32
#define NGT       50
#define NGTP      64         // gts padded to 4 WMMA tiles of 16
#define HH        64
#define WW        64
#define HW        (HH*WW)    // 4096
#define KA        (AB*HW)    // 36864
#define IMGSZ     1024.0f
#define RPN_NEGT  0.3f
#define RPN_POST  0.7f
#define RPN_BSZ   256
#define NUM_FG    128

typedef unsigned int v4u __attribute__((ext_vector_type(4)));
typedef int          v8i __attribute__((ext_vector_type(8)));
typedef int          v4i __attribute__((ext_vector_type(4)));
typedef float        v2f __attribute__((ext_vector_type(2)));
typedef float        v8f __attribute__((ext_vector_type(8)));

// ---------------- CDNA5 Tensor Data Mover: 1-D dword copy global -> LDS -----
// D# packed per CDNA5 ISA 8.3/8.4. 6-arg builtin (clang-23/therock form):
//   (uint32x4 g0, int32x8 g1, int32x4 g2, int32x4 g3, int32x8 g4, i32 cpol)
__device__ __forceinline__ void tdm_load_dwords(void* lds_dst, const void* gsrc, unsigned n) {
  unsigned long long la = (unsigned long long)lds_dst;  // low32 = LDS offset
  unsigned long long ga = (unsigned long long)gsrc;
  v4u g0 = { 1u,                                  // count=1, user D#
             (unsigned)la,                        // lds_addr
             (unsigned)ga,                        // global_addr[31:0]
             (unsigned)((ga >> 32) & 0x1ffffffull) | (2u << 30) }; // [56:32] | type=2
  v8i g1 = { (int)(2u << 16),                     // wg_mask=0, data_size=4B
             (int)((n & 0xffffu) << 16),          // tensor_dim0[15:0] @ bits63:48
             (int)((n >> 16) | (1u << 16)),       // tensor_dim0[31:16], tensor_dim1=1 lo
             (int)(n << 16),                      // tensor_dim1 hi=0, tile_dim0=n
             1,                                   // tile_dim1=1, tile_dim2=0
             (int)n,                              // tensor_dim0_stride lo32
             0, 0 };
  v4i z4 = {0, 0, 0, 0};
  v8i z8 = {0, 0, 0, 0, 0, 0, 0, 0};
  __builtin_amdgcn_tensor_load_to_lds(g0, g1, z4, z4, z8, 0);
}

__device__ __forceinline__ unsigned hash32(unsigned x) {
  x ^= x >> 16; x *= 0x7feb352du;
  x ^= x >> 15; x *= 0x846ca68bu;
  x ^= x >> 16; return x;
}

// Base anchor a at grid cell (w,h).  np.round == rintf (RNE).
// contract(off): all kernels must produce bitwise-identical anchor coords.
__device__ __forceinline__ void anchor_box(int a, int w, int h,
                                           float& x1, float& y1, float& x2, float& y2) {
#pragma clang fp contract(off)
  int ri = a / 3, si = a % 3;
  float ratio = (ri == 0) ? 0.5f : (ri == 1) ? 1.0f : 2.0f;
  float scale = (si == 0) ? 8.0f : (si == 1) ? 16.0f : 32.0f;
  float ws = rintf(sqrtf(256.0f / ratio));
  float hs = rintf(ws * ratio);
  float bw = ws * scale, bh = hs * scale;
  float sx = (float)w * 16.0f, sy = (float)h * 16.0f;
  x1 = sx + 7.5f - 0.5f * (bw - 1.0f);
  y1 = sy + 7.5f - 0.5f * (bh - 1.0f);
  x2 = sx + 7.5f + 0.5f * (bw - 1.0f);
  y2 = sy + 7.5f + 0.5f * (bh - 1.0f);
}

// +1-pixel-convention IoU. Contraction disabled so all passes produce
// bitwise-identical values (needed for the ov == gt_max hit test). The WMMA
// path computes union base area_a+area_g as area_a*1 + 1*area_g (+0s), whose
// single inexact RNE add is bitwise equal to this v_add_f32.
__device__ __forceinline__ float iou_pair(float ax1, float ay1, float ax2, float ay2,
                                          float gx1, float gy1, float gx2, float gy2) {
#pragma clang fp contract(off)
  float area_a = (ax2 - ax1 + 1.0f) * (ay2 - ay1 + 1.0f);
  float area_g = (gx2 - gx1 + 1.0f) * (gy2 - gy1 + 1.0f);
  float iw = fminf(ax2, gx2) - fmaxf(ax1, gx1) + 1.0f;
  float ih = fminf(ay2, gy2) - fmaxf(ay1, gy1) + 1.0f;
  iw = fmaxf(iw, 0.0f);
  ih = fmaxf(ih, 0.0f);
  float inter = iw * ih;
  return inter / (area_a + area_g - inter);
}

// ---------------- pass 0: zero gt_max ---------------------------------------
__global__ void k_init(unsigned* __restrict__ gtmax) {
  int i = blockIdx.x * blockDim.x + threadIdx.x;
  if (i < BATCH * NGT) gtmax[i] = 0u;
}

// ---------------- pass 1: per-gt max IoU (WMMA-tiled) -----------------------
// Wave handles a 16-anchor strip vs 4 tiles of 16 gts. One
// V_WMMA_F32_16X16X4_F32 per tile computes S[M][N] = area_a[M] + area_g[N]
// (A row = [area_a,1,0,0], B col = [1,area_g,0,0]^T). Per the 16x16 f32 C/D
// layout, lane l / vgpr v holds (M = v + 8*(l>=16), N = l%16); the min/max
// intersection runs on VALU and co-executes with the matrix pipe.
__global__ void __launch_bounds__(256) k_gtmax(const float* __restrict__ gt,
                                               unsigned* __restrict__ gtmax) {
  __shared__ __align__(16) float sgt[NGTP * 4];      // padded gt boxes
  __shared__ float sga[NGTP];                        // gt areas
  __shared__ __align__(16) float sanch[8 * 16 * 4];  // per-wave anchor boxes
  __shared__ float sins[8 * 16];                     // inside flags
  __shared__ unsigned sgmax[NGTP];                   // per-block col maxes

  int b = blockIdx.y;
  int tid = threadIdx.x;
  tdm_load_dwords(sgt, gt + (size_t)b * NGT * 4, NGT * 4);
  // pad gts 50..63 with degenerate (0,0,-1,-1): area 0, IoU 0 vs any anchor
  if (tid >= NGT * 4 && tid < NGTP * 4) sgt[tid] = ((tid & 3) >= 2) ? -1.0f : 0.0f;
  if (tid < NGTP) sgmax[tid] = 0u;
  __builtin_amdgcn_s_wait_tensorcnt(0);
  __syncthreads();

  if (tid < NGTP) {
#pragma clang fp contract(off)
    float gx1 = sgt[tid * 4 + 0], gy1 = sgt[tid * 4 + 1];
    float gx2 = sgt[tid * 4 + 2], gy2 = sgt[tid * 4 + 3];
    sga[tid] = (gx2 - gx1 + 1.0f) * (gy2 - gy1 + 1.0f);
  }

  int wv = tid >> 5, lane = tid & 31;
  int half = lane >> 4, ln = lane & 15;
  int i0 = (blockIdx.x * 8 + wv) * 16;               // wave's anchor base

  float areaA = 0.0f;
  if (half == 0) {
#pragma clang fp contract(off)
    int idx = i0 + ln;                               // output order [a][hw]
    int a = idx / HW, hw = idx % HW;
    int h = hw / WW, w = hw % WW;
    float x1, y1, x2, y2;
    anchor_box(a, w, h, x1, y1, x2, y2);
    bool inside = (x1 >= 0.0f) && (y1 >= 0.0f) && (x2 < IMGSZ) && (y2 < IMGSZ);
    ((float4*)sanch)[wv * 16 + ln] = make_float4(x1, y1, x2, y2);
    sins[wv * 16 + ln] = inside ? 1.0f : 0.0f;
    areaA = (x2 - x1 + 1.0f) * (y2 - y1 + 1.0f);
  }
  __syncthreads();

  // A fragment (16x4 f32): vgpr0 = K0 (lanes<16) / K2 (lanes>=16), vgpr1 = K1/K3
  v2f amat;
  amat.x = areaA;                                    // K0 = area_a, K2 = 0
  amat.y = (half == 0) ? 1.0f : 0.0f;                // K1 = 1,      K3 = 0

  for (int t = 0; t < 4; ++t) {
    // B fragment (4x16 f32): vgpr0 = row K0/K2, vgpr1 = row K1/K3
    v2f bmat;
    bmat.x = (half == 0) ? 1.0f : 0.0f;              // row K0 = ones
    bmat.y = (half == 0) ? sga[t * 16 + ln] : 0.0f;  // row K1 = area_g
    v8f c = {0.0f, 0.0f, 0.0f, 0.0f, 0.0f, 0.0f, 0.0f, 0.0f};
    c = __builtin_amdgcn_wmma_f32_16x16x4_f32(false, amat, false, bmat,
                                              (short)0, c, false, false);
    int g = t * 16 + ln;
    float4 gb = ((const float4*)sgt)[g];
    float colmax = 0.0f;
#pragma unroll
    for (int v = 0; v < 8; ++v) {
      int M = v + 8 * half;
      float4 ab = ((const float4*)sanch)[wv * 16 + M];
      float iou;
      {
#pragma clang fp contract(off)
        float iw = fminf(ab.z, gb.z) - fmaxf(ab.x, gb.x) + 1.0f;
        float ih = fminf(ab.w, gb.w) - fmaxf(ab.y, gb.y) + 1.0f;
        iw = fmaxf(iw, 0.0f);
        ih = fmaxf(ih, 0.0f);
        float inter = iw * ih;
        iou = inter / (c[v] - inter);                // c[v] = area_a + area_g
      }
      iou = (sins[wv * 16 + M] != 0.0f) ? iou : 0.0f;
      colmax = fmaxf(colmax, iou);
    }
    // combine the two half-wave row groups, then per-gt LDS atomic max
    colmax = fmaxf(colmax, __shfl_xor(colmax, 16, 32));
    if (half == 0 && g < NGT)
      atomicMax(&sgmax[g], __float_as_uint(colmax));
  }
  __syncthreads();
  if (tid < NGT) atomicMax(&gtmax[b * NGT + tid], sgmax[tid]);
}

// ---------------- pass 2: labels + subsample priorities ---------------------
__global__ void k_labels(const float* __restrict__ gt, const unsigned* __restrict__ gtmax,
                         float* __restrict__ out_lab, unsigned* __restrict__ prio) {
  __shared__ __align__(16) float sgt[NGT * 4];
  __shared__ float sgm[NGT];
  int b = blockIdx.y;
  tdm_load_dwords(sgt, gt + (size_t)b * NGT * 4, NGT * 4);
  if (threadIdx.x < NGT) sgm[threadIdx.x] = __uint_as_float(gtmax[b * NGT + threadIdx.x]);
  __builtin_amdgcn_s_wait_tensorcnt(0);
  __syncthreads();

  int idx = blockIdx.x * blockDim.x + threadIdx.x;   // output order: [a][hw]
  int a = idx / HW, hw = idx % HW;
  int h = hw / WW, w = hw % WW;
  float ax1, ay1, ax2, ay2;
  anchor_box(a, w, h, ax1, ay1, ax2, ay2);
  bool inside = (ax1 >= 0.0f) && (ay1 >= 0.0f) && (ax2 < IMGSZ) && (ay2 < IMGSZ);

  float label = -1.0f;
  if (inside) {
    float best = -1.0f;
    bool hit = false;
    for (int g = 0; g < NGT; ++g) {
      float4 gb = ((const float4*)sgt)[g];
      float v = iou_pair(ax1, ay1, ax2, ay2, gb.x, gb.y, gb.z, gb.w);
      if (v > best) best = v;
      float gm = sgm[g];
      hit = hit || ((v == gm) && (gm > 0.0f));        // bitwise match by construction
    }
    if (best < RPN_NEGT) label = 0.0f;
    if (best >= RPN_POST) label = 1.0f;
    if (hit) label = 1.0f;
  }
  out_lab[(size_t)b * KA + idx] = label;
  // Deterministic per-(image,anchor,class) priority hash. NOTE: reference uses
  // jax threefry; unverifiable here, same rank-subsampling semantics.
  unsigned ph = (label == 1.0f) ? 0x9e3779b9u : 0x85ebca6bu;
  prio[(size_t)b * KA + idx] = hash32(((unsigned)(b * KA) + (unsigned)(hw * AB + a)) ^ ph);
}

// ---------------- pass 3: exact-k fg/bg subsample (one block per image) -----
__global__ void __launch_bounds__(1024) k_subsample(float* __restrict__ out_lab,
                                                    const unsigned* __restrict__ prio) {
  __shared__ unsigned hist[4096];
  __shared__ unsigned cand[1024];
  __shared__ unsigned scal[8];
  int b = blockIdx.x;
  int tid = threadIdx.x;
  float* lab = out_lab + (size_t)b * KA;
  const unsigned* pr = prio + (size_t)b * KA;

  int kept_fg = 0;
  for (int phase = 0; phase < 2; ++phase) {
    float want = (phase == 0) ? 1.0f : 0.0f;
    int k = (phase == 0) ? NUM_FG : (RPN_BSZ - kept_fg);

    for (int j = tid; j < 4096; j += 1024) hist[j] = 0u;
    if (tid == 0) { scal[3] = 0u; scal[4] = 0xffffffffu; }
    __syncthreads();

    for (int idx = tid; idx < KA; idx += 1024)
      if (lab[idx] == want) atomicAdd(&hist[pr[idx] >> 20], 1u);
    __syncthreads();

    if (tid == 0) {
      unsigned tot = 0;
      for (int j = 0; j < 4096; ++j) tot += hist[j];
      scal[0] = tot;
      if (tot > (unsigned)k) {
        unsigned cum = 0, tb = 0, cb = 0;
        for (int j = 0; j < 4096; ++j) {
          if (cum + hist[j] >= (unsigned)k) { tb = (unsigned)j; cb = cum; break; }
          cum += hist[j];
        }
        scal[1] = tb; scal[2] = cb;
      }
    }
    __syncthreads();

    unsigned tot = scal[0];
    if ((int)tot > k) {
      unsigned tb = scal[1], cb = scal[2];
      // gather boundary-bin candidates
      for (int idx = tid; idx < KA; idx += 1024) {
        if (lab[idx] == want) {
          unsigned p = pr[idx];
          if ((p >> 20) == tb) {
            unsigned pos = atomicAdd(&scal[3], 1u);
            if (pos < 1024u) cand[pos] = p;
          }
        }
      }
      __syncthreads();
      int m = (int)(scal[3] < 1024u ? scal[3] : 1024u);
      int r = k - (int)cb;                 // need r smallest from boundary bin
      if (tid < m) {                       // parallel rank selection
        unsigned v = cand[tid];
        int lt = 0, eq = 0;
        for (int j = 0; j < m; ++j) { lt += (cand[j] < v); eq += (cand[j] == v); }
        if (lt <= r - 1 && r - 1 < lt + eq) scal[4] = v;
      }
      __syncthreads();
      unsigned T = scal[4];
      for (int idx = tid; idx < KA; idx += 1024) {
        if (lab[idx] == want) {
          unsigned p = pr[idx];
          bool keep = ((p >> 20) < tb) || (((p >> 20) == tb) && (p <= T));
          if (!keep) lab[idx] = -1.0f;
        }
      }
    }
    if (phase == 0) kept_fg = (int)(tot < (unsigned)NUM_FG ? tot : (unsigned)NUM_FG);
    __syncthreads();
  }
}

// ---------------- pass 4: regression targets, output-channel order ----------
__global__ void k_reg(const float* __restrict__ gt, const float* __restrict__ out_lab,
                      float* __restrict__ out_reg) {
  __shared__ __align__(16) float sgt[NGT * 4];
  int b = blockIdx.y;
  tdm_load_dwords(sgt, gt + (size_t)b * NGT * 4, NGT * 4);
  __builtin_amdgcn_s_wait_tensorcnt(0);
  __syncthreads();

  int idx = blockIdx.x * blockDim.x + threadIdx.x;   // over [4A*HW], coalesced
  int c = idx / HW, hw = idx % HW;
  int a = c >> 2, j = c & 3;
  float label = out_lab[(size_t)b * KA + a * HW + hw];
  float val = 0.0f;
  if (label == 1.0f) {
    int h = hw / WW, w = hw % WW;
    float ax1, ay1, ax2, ay2;
    anchor_box(a, w, h, ax1, ay1, ax2, ay2);
    float best = -1.0f; int arg = 0;
    for (int g = 0; g < NGT; ++g) {
      float4 gb = ((const float4*)sgt)[g];
      float v = iou_pair(ax1, ay1, ax2, ay2, gb.x, gb.y, gb.z, gb.w);
      if (v > best) { best = v; arg = g; }
    }
    float4 gb = ((const float4*)sgt)[arg];
    {
#pragma clang fp contract(off)
      float aw = ax2 - ax1 + 1.0f, ah = ay2 - ay1 + 1.0f;
      float acx = ax1 + 0.5f * (aw - 1.0f), acy = ay1 + 0.5f * (ah - 1.0f);
      float gw = gb.z - gb.x + 1.0f, gh = gb.w - gb.y + 1.0f;
      float gcx = gb.x + 0.5f * (gw - 1.0f), gcy = gb.y + 0.5f * (gh - 1.0f);
      val = (j == 0) ? (gcx - acx) / aw
          : (j == 1) ? (gcy - acy) / ah
          : (j == 2) ? logf(gw / aw)
                     : logf(gh / ah);
    }
  }
  out_reg[(size_t)b * 4 * KA + idx] = val;
}

extern "C" void kernel_launch(void* const* d_in, const int* in_sizes, int n_in,
                              void* d_out, int out_size, void* d_ws, size_t ws_size,
                              hipStream_t stream) {
  (void)in_sizes; (void)n_in; (void)out_size; (void)ws_size;
  const float* gt = (const float*)d_in[0];           // [B*G, 4] fp32
  float* out_lab = (float*)d_out;                    // [B, 9, 64, 64]
  float* out_reg = out_lab + (size_t)BATCH * KA;     // [B, 36, 64, 64]
  unsigned* gtmax = (unsigned*)d_ws;                 // B*G floats (as bits)
  unsigned* prio  = (unsigned*)((char*)d_ws + 8192); // B*KA uints

  k_init     <<<dim3((BATCH * NGT + 255) / 256), dim3(256), 0, stream>>>(gtmax);
  k_gtmax    <<<dim3(KA / 128, BATCH),           dim3(256), 0, stream>>>(gt, gtmax);
  k_labels   <<<dim3(KA / 256, BATCH),           dim3(256), 0, stream>>>(gt, gtmax, out_lab, prio);
  k_subsample<<<dim3(BATCH),                     dim3(1024), 0, stream>>>(out_lab, prio);
  k_reg      <<<dim3(4 * KA / 256, BATCH),       dim3(256), 0, stream>>>(gt, out_lab, out_reg);
}